// NormalizedDynamicsSmartK_57561151701125
// MI455X (gfx1250) — compile-verified
//
#include <hip/hip_runtime.h>
#include <math.h>
#include <stdint.h>

#define NPTS 8192
#define KSEL 32
#define FEPS 1e-8f

typedef float v2f __attribute__((ext_vector_type(2)));
typedef float v8f __attribute__((ext_vector_type(8)));

// ---- workspace layout (float offsets) ----
#define WS_MEAN   0            // 2 floats
#define WS_STD    2            // 2 floats
#define WS_SCALE  4            // 2 floats
#define WS_INVDEN 8            // NPTS floats: 1/(2*sigma^2+eps)
#define WS_H      (8 + NPTS)   // 2*NPTS floats: h before re-scaling

// =============== threefry2x32-20 (JAX-compatible) ===============
__device__ __forceinline__ uint32_t rotl32(uint32_t v, int n) {
  return (v << n) | (v >> (32 - n));
}
__device__ __forceinline__ void threefry2x32(uint32_t k0, uint32_t k1,
                                             uint32_t c0, uint32_t c1,
                                             uint32_t& o0, uint32_t& o1) {
  uint32_t ks0 = k0, ks1 = k1, ks2 = 0x1BD11BDAu ^ k0 ^ k1;
  uint32_t x0 = c0 + ks0, x1 = c1 + ks1;
#define TF_R(a,b,cc,d) \
  x0 += x1; x1 = rotl32(x1,a); x1 ^= x0; \
  x0 += x1; x1 = rotl32(x1,b); x1 ^= x0; \
  x0 += x1; x1 = rotl32(x1,cc); x1 ^= x0; \
  x0 += x1; x1 = rotl32(x1,d); x1 ^= x0;
  TF_R(13,15,26,6)  x0 += ks1; x1 += ks2 + 1u;
  TF_R(17,29,16,24) x0 += ks2; x1 += ks0 + 2u;
  TF_R(13,15,26,6)  x0 += ks0; x1 += ks1 + 3u;
  TF_R(17,29,16,24) x0 += ks1; x1 += ks2 + 4u;
  TF_R(13,15,26,6)  x0 += ks2; x1 += ks0 + 5u;
#undef TF_R
  o0 = x0; o1 = x1;
}

// bits for flat index f of jax.random.normal(key(42), (NPTS,2)):
// counts = iota(2*NPTS); halves (c, c+NPTS); out = concat(x0_out, x1_out)
__device__ __forceinline__ uint32_t jax_bits(uint32_t f) {
  uint32_t o0, o1;
  if (f < (uint32_t)NPTS) { threefry2x32(0u, 42u, f, f + NPTS, o0, o1); return o0; }
  threefry2x32(0u, 42u, f - NPTS, f, o0, o1); return o1;
}
__device__ __forceinline__ float jax_normal_from_bits(uint32_t bits) {
  float f = __uint_as_float((bits >> 9) | 0x3f800000u) - 1.0f;  // [0,1)
  const float lo = -0.99999994f;                                 // nextafter(-1,0)
  float u = fmaf(f, 1.0f - lo, lo);
  u = fmaxf(lo, u);
  return 1.4142135623730951f * erfinvf(u);
}

// =============== async global->LDS copy (CDNA5 async path) ===============
// GV mode: per-lane 64-bit global address, LDS offset in VDST VGPR.
// Generic shared pointer low 32 bits == LDS offset (ISA 10.2 aperture map).
__device__ __forceinline__ void async_copy_to_lds(float* lds, const float* g,
                                                  int bytes) {
  unsigned lbase = (unsigned)(size_t)(void*)lds;
  unsigned long long gbase = (unsigned long long)(size_t)(const void*)g;
  int nchunk = bytes >> 4;  // 16B per lane-op
  for (int c = threadIdx.x; c < nchunk; c += blockDim.x) {
    unsigned la = lbase + ((unsigned)c << 4);
    unsigned long long ga = gbase + ((unsigned long long)c << 4);
    asm volatile("global_load_async_to_lds_b128 %0, %1, off"
                 :: "v"(la), "v"(ga) : "memory");
  }
  asm volatile("s_wait_asynccnt 0" ::: "memory");
  __syncthreads();
}

// =============== kernel 1: mean / std (ddof=1) of x ===============
__global__ __launch_bounds__(1024) void k_stats(const float* __restrict__ x,
                                                float* __restrict__ ws) {
  __shared__ float4 red[1024];
  float s0 = 0.f, s1 = 0.f, q0 = 0.f, q1 = 0.f;
  for (int i = threadIdx.x; i < NPTS; i += 1024) {
    float a = x[2 * i], b = x[2 * i + 1];
    s0 += a; s1 += b; q0 = fmaf(a, a, q0); q1 = fmaf(b, b, q1);
  }
  red[threadIdx.x] = make_float4(s0, s1, q0, q1);
  __syncthreads();
  for (int s = 512; s > 0; s >>= 1) {
    if ((int)threadIdx.x < s) {
      float4 m = red[threadIdx.x], o = red[threadIdx.x + s];
      red[threadIdx.x] = make_float4(m.x + o.x, m.y + o.y, m.z + o.z, m.w + o.w);
    }
    __syncthreads();
  }
  if (threadIdx.x == 0) {
    float4 t = red[0];
    float n = (float)NPTS;
    float m0 = t.x / n, m1 = t.y / n;
    float v0 = (t.z - n * m0 * m0) / (n - 1.0f);
    float v1 = (t.w - n * m1 * m1) / (n - 1.0f);
    ws[WS_MEAN] = m0; ws[WS_MEAN + 1] = m1;
    ws[WS_STD] = sqrtf(fmaxf(v0, 0.f)); ws[WS_STD + 1] = sqrtf(fmaxf(v1, 0.f));
  }
}

// =============== kernel 2: per-row 32-NN radius -> invden ===============
// Distances are mean-invariant, so raw coords are fine.
__global__ __launch_bounds__(256) void k_sigma(const float* __restrict__ x,
                                               float* __restrict__ ws) {
  __shared__ float s_xy[2 * NPTS];  // 64 KB, all points
  async_copy_to_lds(s_xy, x, 2 * NPTS * 4);

  int i = blockIdx.x * 256 + threadIdx.x;
  const float2* p = (const float2*)s_xy;
  float2 pi = p[i];

  float topk[KSEL];
#pragma unroll
  for (int t = 0; t < KSEL; ++t) topk[t] = 3.0e38f;
  float kmax = 3.0e38f;

  for (int j = 0; j < NPTS; ++j) {
    float2 pj = p[j];
    float dx = pi.x - pj.x, dy = pi.y - pj.y;
    float d2 = fmaf(dx, dx, dy * dy);
    if (d2 < kmax) {  // replace one max entry, recompute max
      bool done = false;
#pragma unroll
      for (int t = 0; t < KSEL; ++t) {
        bool hit = (topk[t] == kmax) && !done;
        topk[t] = hit ? d2 : topk[t];
        done = done || hit;
      }
      kmax = topk[0];
#pragma unroll
      for (int t = 1; t < KSEL; ++t) kmax = fmaxf(kmax, topk[t]);
    }
  }
  // kmax == sigma^2 (squared kth-smallest distance, self included)
  ws[WS_INVDEN + i] = 1.0f / (2.0f * kmax + FEPS);
}

// =============== kernel 3: WMMA distance tiles -> drift -> h ===============
// d2 = A x B with A row i = [x_i, y_i, sq_i, 1], B col j = [-2x_j, -2y_j, 1, sq_j]
// One v_wmma_f32_16x16x4_f32 per 16x16 tile. Each wave owns 16 rows.
__global__ __launch_bounds__(256) void k_drift(const float* __restrict__ x,
                                               float* __restrict__ ws) {
  __shared__ float s_xy[2 * NPTS];  // 64 KB
  async_copy_to_lds(s_xy, x, 2 * NPTS * 4);

  const int lane = threadIdx.x & 31;
  const int wave = threadIdx.x >> 5;
  const int hv = lane >> 4;      // C-layout lane half
  const int l15 = lane & 15;
  const int rowbase = (blockIdx.x * 8 + wave) * 16;
  const int row_off = rowbase + 8 * hv;  // rows held by this lane half

  const float m0 = ws[WS_MEAN], m1 = ws[WS_MEAN + 1];

  // A fragment (ISA 7.12.2, 32-bit A 16x4): vgpr0 = K0|K2, vgpr1 = K1|K3
  int ar = rowbase + l15;
  float axr = s_xy[2 * ar], ayr = s_xy[2 * ar + 1];
  v2f A;
  if (hv == 0) { A.x = axr; A.y = ayr; }                       // K=0: x, K=1: y
  else         { A.x = fmaf(axr, axr, ayr * ayr); A.y = 1.f; } // K=2: sq, K=3: 1

  // pre-scale 1/(2s^2+eps) by -log2(e): w = exp2(d * ninv)  (one mul + v_exp_f32)
  float ninv[8];
#pragma unroll
  for (int r = 0; r < 8; ++r)
    ninv[r] = ws[WS_INVDEN + row_off + r] * -1.4426950408889634f;

  float accS[8], accX[8], accY[8];
#pragma unroll
  for (int r = 0; r < 8; ++r) { accS[r] = 0.f; accX[r] = 0.f; accY[r] = 0.f; }

#pragma unroll 2
  for (int t = 0; t < NPTS / 16; ++t) {
    int cj = t * 16 + l15;
    float xj = s_xy[2 * cj], yj = s_xy[2 * cj + 1];
    v2f B;
    if (hv == 0) { B.x = -2.0f * xj; B.y = -2.0f * yj; }           // K=0,1
    else         { B.x = 1.0f; B.y = fmaf(xj, xj, yj * yj); }      // K=2,3
    v8f C = {0.f, 0.f, 0.f, 0.f, 0.f, 0.f, 0.f, 0.f};
    C = __builtin_amdgcn_wmma_f32_16x16x4_f32(false, A, false, B,
                                              (short)0, C, false, false);
#pragma unroll
    for (int r = 0; r < 8; ++r) {
      // |d2|: fabs folds into v_sqrt source modifier; negatives only arise
      // from rounding noise at the (near-)diagonal where true d^2 ~ 0.
      float d = __builtin_amdgcn_sqrtf(__builtin_fabsf(C[r]));  // v_sqrt_f32 |v|
      float w = __builtin_amdgcn_exp2f(d * ninv[r]);            // v_exp_f32
      accS[r] += w;
      accX[r] = fmaf(w, xj, accX[r]);
      accY[r] = fmaf(w, yj, accY[r]);
    }
  }

  // reduce across the 16 lanes of each half (masks < 16 stay within half)
#pragma unroll
  for (int r = 0; r < 8; ++r) {
#pragma unroll
    for (int m = 1; m < 16; m <<= 1) {
      accS[r] += __shfl_xor(accS[r], m, 32);
      accX[r] += __shfl_xor(accX[r], m, 32);
      accY[r] += __shfl_xor(accY[r], m, 32);
    }
  }

  if (l15 == 0) {  // lanes 0 and 16 finalize their 8 rows each
    for (int r = 0; r < 8; ++r) {
      int i = row_off + r;
      float S = accS[r], X = accX[r], Y = accY[r];
      float xc0 = s_xy[2 * i] - m0, xc1 = s_xy[2 * i + 1] - m1;
      float dn = 1.0f / (S + FEPS);
      float dr0 = (X - m0 * S) * dn;  // centered drift
      float dr1 = (Y - m1 * S) * dn;
      float n0 = jax_normal_from_bits(jax_bits(2u * (uint32_t)i)) * 0.01f;
      float n1 = jax_normal_from_bits(jax_bits(2u * (uint32_t)i + 1u)) * 0.01f;
      // h = xc + 0.5*(drift - xc) + noise
      ws[WS_H + 2 * i]     = 0.5f * (xc0 + dr0) + n0;
      ws[WS_H + 2 * i + 1] = 0.5f * (xc1 + dr1) + n1;
    }
  }
}

// =============== kernel 4: std(h) -> scale factors ===============
__global__ __launch_bounds__(1024) void k_scale(float* __restrict__ ws) {
  __shared__ float4 red[1024];
  float s0 = 0.f, s1 = 0.f, q0 = 0.f, q1 = 0.f;
  for (int i = threadIdx.x; i < NPTS; i += 1024) {
    float a = ws[WS_H + 2 * i], b = ws[WS_H + 2 * i + 1];
    s0 += a; s1 += b; q0 = fmaf(a, a, q0); q1 = fmaf(b, b, q1);
  }
  red[threadIdx.x] = make_float4(s0, s1, q0, q1);
  __syncthreads();
  for (int s = 512; s > 0; s >>= 1) {
    if ((int)threadIdx.x < s) {
      float4 m = red[threadIdx.x], o = red[threadIdx.x + s];
      red[threadIdx.x] = make_float4(m.x + o.x, m.y + o.y, m.z + o.z, m.w + o.w);
    }
    __syncthreads();
  }
  if (threadIdx.x == 0) {
    float4 t = red[0];
    float n = (float)NPTS;
    float m0 = t.x / n, m1 = t.y / n;
    float v0 = (t.z - n * m0 * m0) / (n - 1.0f);
    float v1 = (t.w - n * m1 * m1) / (n - 1.0f);
    float cs0 = sqrtf(fmaxf(v0, 0.f)), cs1 = sqrtf(fmaxf(v1, 0.f));
    ws[WS_SCALE]     = ws[WS_STD]     / (cs0 + FEPS);
    ws[WS_SCALE + 1] = ws[WS_STD + 1] / (cs1 + FEPS);
  }
}

// =============== kernel 5: out = h*scale + mean ===============
__global__ __launch_bounds__(256) void k_apply(const float* __restrict__ ws,
                                               float* __restrict__ out) {
  int e = blockIdx.x * 256 + threadIdx.x;  // 0 .. 2*NPTS-1
  int d = e & 1;
  out[e] = fmaf(ws[WS_H + e], ws[WS_SCALE + d], ws[WS_MEAN + d]);
}

extern "C" void kernel_launch(void* const* d_in, const int* in_sizes, int n_in,
                              void* d_out, int out_size, void* d_ws, size_t ws_size,
                              hipStream_t stream) {
  (void)in_sizes; (void)n_in; (void)out_size; (void)ws_size;
  const float* x = (const float*)d_in[0];
  float* ws = (float*)d_ws;
  float* out = (float*)d_out;

  k_stats<<<1, 1024, 0, stream>>>(x, ws);
  k_sigma<<<NPTS / 256, 256, 0, stream>>>(x, ws);
  k_drift<<<NPTS / 128, 256, 0, stream>>>(x, ws);   // 8 waves x 16 rows per WG
  k_scale<<<1, 1024, 0, stream>>>(ws);
  k_apply<<<(2 * NPTS) / 256, 256, 0, stream>>>(ws, out);
}